// GIPEPS_41120016892163
// MI455X (gfx1250) — compile-verified
//
#include <hip/hip_runtime.h>
#include <stdint.h>

// ---------------------------------------------------------------------------
// Exact transfer-matrix contraction of the 4x4 PEPS sample amplitude.
// All bonds zero-padded to 8 (exact: padded entries are 0, contribute 0).
//   M_rc[u][d][l][r]                       (16 x 4096 f32)
//   T_c[d][(l0,l1)][(r0,r1)] = sum_m M0[0,m,l0,r0]*M1[m,d,l1,r1]   (rows 0,1)
//   B_c[d][(l2,l3)][(r2,r3)] = sum_m M2[d,m,l2,r2]*M3[m,0,l3,r3]   (rows 2,3)
//   F_c[R,R'] = sum_{L,L',d} F_{c-1}[L,L'] T_c[L,d,R] B_c[L',d,R']
//             = sum_d (T_d^T * F_{c-1} * B_d)      -> amplitude = F_3[0,0]
// ---------------------------------------------------------------------------

typedef __attribute__((ext_vector_type(2))) float        v2f;
typedef __attribute__((ext_vector_type(8))) float        v8f;
typedef __attribute__((ext_vector_type(4))) unsigned int v4u;
typedef __attribute__((ext_vector_type(4))) int          v4i;
typedef __attribute__((ext_vector_type(8))) int          v8i;

#if __has_builtin(__builtin_amdgcn_wmma_f32_16x16x4_f32)
#define USE_WMMA 1
#else
#define USE_WMMA 0
#endif

#if __has_builtin(__builtin_amdgcn_tensor_load_to_lds)
#define USE_TDM 1
#else
#define USE_TDM 0
#endif

// ---------------------------------------------------------------------------
// Kernel 1: slice each 9-D site tensor with sampled indices into dense
// zero-padded M[8][8][8][8].
// ---------------------------------------------------------------------------
struct SliceArgs {
    const float* t[16];
    const int*   sample;
    float*       M;
};

__global__ __launch_bounds__(256) void k_slice(SliceArgs a) {
    const int site = blockIdx.x;
    const int r = site >> 2, c = site & 3;
    const int* sm = a.sample;               // [0:16) sites, [16:28) h, [28:40) v
    const int s  = sm[site];
    const int ku = (r > 0) ? sm[28 + (r - 1) * 4 + c] : 0;
    const int kd = (r < 3) ? sm[28 + r * 4 + c]       : 0;
    const int kl = (c > 0) ? sm[16 + r * 3 + (c - 1)] : 0;
    const int kr = (c < 3) ? sm[16 + r * 3 + c]       : 0;

    const int kud = r > 0 ? 4 : 1, mud = r > 0 ? 8 : 1;
    const int kdd = r < 3 ? 4 : 1, mdd = r < 3 ? 8 : 1;
    const int kld = c > 0 ? 4 : 1, mld = c > 0 ? 8 : 1;
    const int krd = c < 3 ? 4 : 1, mrd = c < 3 ? 8 : 1;

    // strides for shape (4, kud, mud, kdd, mdd, kld, mld, krd, mrd), row-major
    const long s_mr = 1;
    const long s_kr = (long)mrd;
    const long s_ml = s_kr * krd;
    const long s_kl = s_ml * mld;
    const long s_md = s_kl * kld;
    const long s_kd = s_md * mdd;
    const long s_mu = s_kd * kdd;
    const long s_ku = s_mu * mud;
    const long s_ph = s_ku * kud;

    const float* base = a.t[site] + s * s_ph + ku * s_ku + kd * s_kd +
                        kl * s_kl + kr * s_kr;
    float* M = a.M + site * 4096;
    for (int i = threadIdx.x; i < 4096; i += blockDim.x) {
        const int u  = i >> 9;
        const int dd = (i >> 6) & 7;
        const int ll = (i >> 3) & 7;
        const int rr = i & 7;
        float v = 0.0f;
        if (u < mud && dd < mdd && ll < mld && rr < mrd)
            v = base[u * s_mu + dd * s_md + ll * s_ml + rr * s_mr];
        M[i] = v;
    }
}

// ---------------------------------------------------------------------------
// Kernel 2: build the 8 two-row MPS tensors T_c / B_c, layout [d][L][R]
// (each d-slice contiguous 16 KB -> clean TDM tile for kernel 3).
// ---------------------------------------------------------------------------
__global__ __launch_bounds__(256) void k_buildTB(const float* __restrict__ M,
                                                 float* __restrict__ TB) {
    const int  o   = blockIdx.x;     // 0..3 top columns, 4..7 bottom columns
    const int  c   = o & 3;
    const bool top = o < 4;
    __shared__ float A[4096];
    __shared__ float B[4096];
    const float* Ma = M + ((top ? 0 : 2) * 4 + c) * 4096;  // row 0 / row 2
    const float* Mb = M + ((top ? 1 : 3) * 4 + c) * 4096;  // row 1 / row 3
    __builtin_prefetch(Ma, 0, 3);
    __builtin_prefetch(Mb, 0, 3);
    for (int i = threadIdx.x; i < 4096; i += blockDim.x) {
        A[i] = Ma[i];
        B[i] = Mb[i];
    }
    __syncthreads();

    float* out = TB + o * 32768;
    for (int i = threadIdx.x; i < 32768; i += blockDim.x) {
        const int d  = i >> 12;
        const int L  = (i >> 6) & 63;
        const int R  = i & 63;
        const int la = L >> 3, lb = L & 7;
        const int ra = R >> 3, rb = R & 7;
        // top:    sum_m A[0][m][la][ra] * B[m][d][lb][rb]
        // bottom: sum_m A[d][m][la][ra] * B[m][0][lb][rb]
        const int ai = (top ? 0 : d) * 512 + la * 8 + ra;
        const int bi = (top ? d * 64 : 0) + lb * 8 + rb;
        float acc = 0.0f;
#pragma unroll
        for (int m = 0; m < 8; ++m) acc += A[ai + m * 64] * B[bi + m * 512];
        out[i] = acc;
    }
}

// ---------------------------------------------------------------------------
// TDM: async 1-D tile (n floats) global -> LDS. Issued once per wave.
// ---------------------------------------------------------------------------
#if USE_TDM
__device__ __forceinline__ void tdm_load_1d(const float* gsrc, float* ldst,
                                            unsigned int n) {
    const unsigned long long ga = (unsigned long long)(uintptr_t)gsrc;
    const unsigned int       la = (unsigned int)(uintptr_t)ldst;  // LDS byte addr
    v4u g0;
    g0.x = 1u;                                         // count=1, user descriptor
    g0.y = la;                                         // lds_addr
    g0.z = (unsigned int)(ga & 0xFFFFFFFFull);         // global_addr[31:0]
    g0.w = (unsigned int)((ga >> 32) & 0x01FFFFFFull)  // global_addr[56:32]
           | (2u << 30);                               // type=2 ("image")
    v8i g1;
    g1[0] = (int)0x00020000u;               // data_size=4B; no multicast/pad
    g1[1] = (int)((n & 0xFFFFu) << 16);     // tensor_dim0[15:0]
    g1[2] = (int)(((n >> 16) & 0xFFFFu) | (1u << 16)); // dim0 hi | tensor_dim1=1
    g1[3] = (int)((n & 0xFFFFu) << 16);     // tile_dim0 = n
    g1[4] = 0;                              // tile_dim1 = tile_dim2 = 0
    g1[5] = (int)n;                         // tensor_dim0_stride
    g1[6] = 0;
    g1[7] = 0;
    v4i z4 = {0, 0, 0, 0};
#if defined(__clang_major__) && (__clang_major__ >= 23)
    v8i z8 = {0, 0, 0, 0, 0, 0, 0, 0};
    __builtin_amdgcn_tensor_load_to_lds(g0, g1, z4, z4, z8, 0);
#else
    __builtin_amdgcn_tensor_load_to_lds(g0, g1, z4, z4, 0);
#endif
}
#endif

// ---------------------------------------------------------------------------
// Kernel 3: single-WGP column sweep. F, H, T_d, B_d live in LDS (64 KB).
// 16 wave32 waves each own one 16x16 tile of the 64x64 matrices.
// ---------------------------------------------------------------------------
__global__ __launch_bounds__(512) void k_contract(const float* __restrict__ TB,
                                                  float* __restrict__ out) {
    __shared__ float Fs[4096];
    __shared__ float Hs[4096];
    __shared__ float Td[4096];
    __shared__ float Bd[4096];

    const int tid  = threadIdx.x;
    const int lane = tid & 31;
    const int wave = tid >> 5;        // 0..15
    const int half = lane >> 4;       // 0/1
    const int ml   = lane & 15;
    const int ti   = (wave >> 2) * 16;  // tile row base
    const int tj   = (wave & 3) * 16;   // tile col base

    for (int i = tid; i < 4096; i += 512) Fs[i] = (i == 0) ? 1.0f : 0.0f;
    __syncthreads();

    for (int c = 0; c < 4; ++c) {
#if USE_WMMA
        v8f acc = {};                 // F_new tile accumulator (registers)
#else
        float acc[8];
#pragma unroll
        for (int v = 0; v < 8; ++v) acc[v] = 0.0f;
#endif
        for (int d = 0; d < 8; ++d) {
            const float* srcT = TB + (c * 8 + d) * 4096;
            const float* srcB = TB + ((4 + c) * 8 + d) * 4096;
#if USE_TDM
            if (wave == 0) {
                tdm_load_1d(srcT, Td, 4096u);
                tdm_load_1d(srcB, Bd, 4096u);
#if __has_builtin(__builtin_amdgcn_s_wait_tensorcnt)
                __builtin_amdgcn_s_wait_tensorcnt(0);
#endif
            }
#else
            for (int i = tid; i < 4096; i += 512) {
                Td[i] = srcT[i];
                Bd[i] = srcB[i];
            }
#endif
            __syncthreads();

            // ---- phase 1: H = Td^T * F  (H[r,l'] = sum_L Td[L,r]*F[L,l'])
#if USE_WMMA
            v8f h = {};
#pragma unroll
            for (int kk = 0; kk < 16; ++kk) {
                const int k0 = kk * 4 + half * 2;
                v2f a, b;
                a.x = Td[(k0 + 0) * 64 + ti + ml];
                a.y = Td[(k0 + 1) * 64 + ti + ml];
                b.x = Fs[(k0 + 0) * 64 + tj + ml];
                b.y = Fs[(k0 + 1) * 64 + tj + ml];
                h = __builtin_amdgcn_wmma_f32_16x16x4_f32(
                    false, a, false, b, (short)0, h, false, false);
            }
#pragma unroll
            for (int v = 0; v < 8; ++v)
                Hs[(ti + half * 8 + v) * 64 + tj + ml] = h[v];
#else
            for (int i = tid; i < 4096; i += 512) {
                const int rr = i >> 6, cc = i & 63;
                float s = 0.0f;
                for (int k = 0; k < 64; ++k) s += Td[k * 64 + rr] * Fs[k * 64 + cc];
                Hs[i] = s;
            }
#endif
            __syncthreads();

            // ---- phase 2: F_new += H * Bd
#if USE_WMMA
#pragma unroll
            for (int kk = 0; kk < 16; ++kk) {
                const int k0 = kk * 4 + half * 2;
                v2f a, b;
                a.x = Hs[(ti + ml) * 64 + k0 + 0];
                a.y = Hs[(ti + ml) * 64 + k0 + 1];
                b.x = Bd[(k0 + 0) * 64 + tj + ml];
                b.y = Bd[(k0 + 1) * 64 + tj + ml];
                acc = __builtin_amdgcn_wmma_f32_16x16x4_f32(
                    false, a, false, b, (short)0, acc, false, false);
            }
#else
            for (int v = 0; v < 8; ++v) {
                const int i = tid + v * 512;
                const int rr = i >> 6, cc = i & 63;
                float s = 0.0f;
                for (int k = 0; k < 64; ++k) s += Hs[rr * 64 + k] * Bd[k * 64 + cc];
                acc[v] += s;
            }
#endif
            __syncthreads();   // protect Hs/Td/Bd before next d overwrites
        }  // d

        // all Fs reads for this column done -> overwrite with F_new
#if USE_WMMA
#pragma unroll
        for (int v = 0; v < 8; ++v)
            Fs[(ti + half * 8 + v) * 64 + tj + ml] = acc[v];
#else
        for (int v = 0; v < 8; ++v) Fs[tid + v * 512] = acc[v];
#endif
        __syncthreads();
    }  // c

    if (tid == 0) out[0] = Fs[0];
}

// ---------------------------------------------------------------------------
// Host launcher (graph-capture safe: only kernel launches on `stream`).
// Workspace layout (floats): [0, 65536)  M (16 x 4096)
//                            [65536, 327680) TB (8 x 32768)   => 1.25 MB
// ---------------------------------------------------------------------------
extern "C" void kernel_launch(void* const* d_in, const int* in_sizes, int n_in,
                              void* d_out, int out_size, void* d_ws,
                              size_t ws_size, hipStream_t stream) {
    (void)in_sizes; (void)n_in; (void)out_size; (void)ws_size;

    float* M  = (float*)d_ws;
    float* TB = M + 16 * 4096;

    SliceArgs sa;
    for (int i = 0; i < 16; ++i) sa.t[i] = (const float*)d_in[i];
    sa.sample = (const int*)d_in[16];
    sa.M      = M;

    k_slice<<<16, 256, 0, stream>>>(sa);
    k_buildTB<<<8, 256, 0, stream>>>(M, TB);
    k_contract<<<1, 512, 0, stream>>>(TB, (float*)d_out);
}